// SetNorm_10840497455442
// MI455X (gfx1250) — compile-verified
//
#include <hip/hip_runtime.h>

typedef __attribute__((ext_vector_type(2))) float v2f;
typedef __attribute__((ext_vector_type(4))) float v4f;
typedef __attribute__((ext_vector_type(8))) float v8f;

// ---------------------------------------------------------------------------
// Pass 1: per-(batch, chunk) partial reduction of sum(x) and sum(x*x).
// V_WMMA_F32_16X16X4_F32 with an all-ones A matrix: D[m][n] = sum_k B[k][n],
// so each WMMA folds 64 input floats into the accumulator. Every row of D is
// the column-sum vector, so sum(all 8 acc VGPRs over all 32 lanes) = 16*total,
// independent of the exact element<->lane mapping.
//
// Loop control is forced scalar (readfirstlane'd wave id -> SGPR trip count),
// so EXEC is provably all-ones for every WMMA and the loop runs on
// s_cmp/s_cbranch with no per-iteration exec masking. Each iteration issues
// two independent global_load_b128 (256 floats/wave) into 8 independent
// accumulators so loads overlap and no WMMA->WMMA RAW exists within an
// iteration. end-i is always a multiple of 256 (n = len*256, chunks rounded
// to 2048 per block = 256 per wave).
// ---------------------------------------------------------------------------
__global__ __launch_bounds__(256) void setnorm_reduce_kernel(
    const float* __restrict__ x, const int* __restrict__ lens,
    float* __restrict__ partials, int SD, int D, int elemsPerBlock)
{
    const int b    = blockIdx.y;
    const int tid  = threadIdx.x;
    // wave id as an SGPR so all derived loop bounds are scalar
    const int wave = __builtin_amdgcn_readfirstlane(tid) >> 5;
    const int lane = tid & 31;

    const int n = lens[b] * D;                    // valid elems; multiple of 256
    const int elemsPerWave = elemsPerBlock >> 3;  // 8 waves; multiple of 256

    const int i   = blockIdx.x * elemsPerBlock + wave * elemsPerWave;
    int end       = i + elemsPerWave;
    if (end > n) end = n;
    const int rem    = end - i;                   // multiple of 256 (or <= 0)
    const int nUnits = rem > 0 ? (rem >> 8) : 0;  // 256-elem units

    v8f aS0 = {}, aS1 = {}, aS2 = {}, aS3 = {};   // sums
    v8f aQ0 = {}, aQ1 = {}, aQ2 = {}, aQ3 = {};   // sums of squares
    const v2f onesA = {1.0f, 1.0f};               // all-ones 16x4 A matrix

    const float* p = x + (long long)b * SD + i + lane * 4;
    for (int u = 0; u < nUnits; ++u, p += 256) {
        v4f va = *(const v4f*)(p);        // two independent b128 loads in flight
        v4f vb = *(const v4f*)(p + 128);
        v2f a01 = {va.x, va.y}, a23 = {va.z, va.w};
        v2f b01 = {vb.x, vb.y}, b23 = {vb.z, vb.w};
        v2f qa01 = a01 * a01, qa23 = a23 * a23;
        v2f qb01 = b01 * b01, qb23 = b23 * b23;
        aS0 = __builtin_amdgcn_wmma_f32_16x16x4_f32(false, onesA, false, a01,  (short)0, aS0, false, false);
        aS1 = __builtin_amdgcn_wmma_f32_16x16x4_f32(false, onesA, false, a23,  (short)0, aS1, false, false);
        aS2 = __builtin_amdgcn_wmma_f32_16x16x4_f32(false, onesA, false, b01,  (short)0, aS2, false, false);
        aS3 = __builtin_amdgcn_wmma_f32_16x16x4_f32(false, onesA, false, b23,  (short)0, aS3, false, false);
        aQ0 = __builtin_amdgcn_wmma_f32_16x16x4_f32(false, onesA, false, qa01, (short)0, aQ0, false, false);
        aQ1 = __builtin_amdgcn_wmma_f32_16x16x4_f32(false, onesA, false, qa23, (short)0, aQ1, false, false);
        aQ2 = __builtin_amdgcn_wmma_f32_16x16x4_f32(false, onesA, false, qb01, (short)0, aQ2, false, false);
        aQ3 = __builtin_amdgcn_wmma_f32_16x16x4_f32(false, onesA, false, qb23, (short)0, aQ3, false, false);
    }

    // Per-lane fold of all accumulator registers (16x over-counted overall).
    float s = 0.f, q = 0.f;
#pragma unroll
    for (int k = 0; k < 8; ++k) {
        s += (aS0[k] + aS1[k]) + (aS2[k] + aS3[k]);
        q += (aQ0[k] + aQ1[k]) + (aQ2[k] + aQ3[k]);
    }

    __shared__ float rs[256];
    __shared__ float rq[256];
    rs[tid] = s; rq[tid] = q;
    __syncthreads();
    for (int off = 128; off > 0; off >>= 1) {
        if (tid < off) { rs[tid] += rs[tid + off]; rq[tid] += rq[tid + off]; }
        __syncthreads();
    }
    if (tid == 0) {
        float* pp = partials + ((long long)b * gridDim.x + blockIdx.x) * 2;
        pp[0] = rs[0] * 0.0625f;   // /16: all 16 rows of D replicate colsums
        pp[1] = rq[0] * 0.0625f;
    }
}

// ---------------------------------------------------------------------------
// Pass 2: finalize per-sample mean / rstd from the block partials.
// var = E[x^2] - mean^2 (masked entries are exact zeros, so this identity
// matches the reference's masked variance).
// ---------------------------------------------------------------------------
__global__ __launch_bounds__(32) void setnorm_stats_kernel(
    const float* __restrict__ partials, const int* __restrict__ lens,
    float* __restrict__ stats, int blocksPerBatch, int D)
{
    const int b = blockIdx.x;
    const int t = threadIdx.x;
    __shared__ float ss[32];
    __shared__ float sq[32];
    float s = 0.f, q = 0.f;
    for (int j = t; j < blocksPerBatch; j += 32) {
        const float* p = partials + ((long long)b * blocksPerBatch + j) * 2;
        s += p[0]; q += p[1];
    }
    ss[t] = s; sq[t] = q;
    __syncthreads();
    if (t == 0) {
        float S = 0.f, Q = 0.f;
        for (int j = 0; j < 32; ++j) { S += ss[j]; Q += sq[j]; }
        float cnt  = fmaxf((float)lens[b] * (float)D, 1.0f);
        float mean = S / cnt;
        float var  = fmaxf(Q / cnt - mean * mean, 0.0f);
        float rstd = __frsqrt_rn(var + 1e-5f);
        stats[b * 2 + 0] = mean;
        stats[b * 2 + 1] = rstd;
    }
}

// ---------------------------------------------------------------------------
// Pass 3: streaming normalize. Input re-read should hit the 192MB L2 (whole
// 134MB tensor fits); output written with non-temporal stores so it does not
// evict the input working set. Padding region written as exact zeros.
// ---------------------------------------------------------------------------
__global__ __launch_bounds__(256) void setnorm_apply_kernel(
    const float* __restrict__ x, const int* __restrict__ lens,
    const float* __restrict__ stats, float* __restrict__ out, int SD, int D)
{
    const int b   = blockIdx.y;
    const int idx = (blockIdx.x * 256 + threadIdx.x) * 4;
    if (idx >= SD) return;
    const long long base = (long long)b * SD + idx;
    const int n = lens[b] * D;   // multiple of 256 -> float4 units fully in/out
    v4f r = {};
    if (idx < n) {
        v4f v = *(const v4f*)(x + base);
        const float mean = stats[b * 2 + 0];
        const float rstd = stats[b * 2 + 1];
        r = (v - mean) * rstd;
    }
    __builtin_nontemporal_store(r, (v4f*)(out + base));
}

// ---------------------------------------------------------------------------
extern "C" void kernel_launch(void* const* d_in, const int* in_sizes, int n_in,
                              void* d_out, int out_size, void* d_ws, size_t ws_size,
                              hipStream_t stream) {
    const float* x    = (const float*)d_in[0];
    const int*   lens = (const int*)d_in[1];
    float*       out  = (float*)d_out;

    const int total = in_sizes[0];     // B*S*D
    const int B     = in_sizes[1];
    const int D     = 256;             // fixed by the reference problem
    const int SD    = total / B;       // S*D = 1,048,576

    const int blocksPerBatch = 32;
    // elems per block: cover SD, rounded to 2048 so per-wave chunks are
    // multiples of 256 (exact 256-elem unit loop, no remainder handling)
    int elemsPerBlock = (SD + blocksPerBatch - 1) / blocksPerBatch;
    elemsPerBlock = (elemsPerBlock + 2047) & ~2047;

    float* partials = (float*)d_ws;                               // B*bpB*2 floats
    float* stats    = partials + (size_t)B * blocksPerBatch * 2;  // B*2 floats

    dim3 g1(blocksPerBatch, B);
    setnorm_reduce_kernel<<<g1, 256, 0, stream>>>(x, lens, partials, SD, D, elemsPerBlock);

    setnorm_stats_kernel<<<B, 32, 0, stream>>>(partials, lens, stats, blocksPerBatch, D);

    dim3 g3((SD + 1023) / 1024, B);
    setnorm_apply_kernel<<<g3, 256, 0, stream>>>(x, lens, stats, out, SD, D);
}